// ACR_73882027426226
// MI455X (gfx1250) — compile-verified
//
#include <hip/hip_runtime.h>
#include <hip/hip_bf16.h>
#include <stdint.h>

// ---------------------------------------------------------------------------
// SCAE template render for MI455X (gfx1250).
// Store-bandwidth-bound: 134 MB out @ 23.3 TB/s => ~5.8us floor.
// CDNA5 paths: TDM tensor_load_to_lds (template tile -> LDS DMA),
// s_wait_tensorcnt, v_tanh_f32 (gfx1250 TRANS op), wave32, b128 stores.
// ---------------------------------------------------------------------------

#define BATCH 128
#define NTMPL 64
#define TS    16
#define IS    64
#define T_ELEMS (NTMPL * TS * TS)          // 16384
#define OUT_PER_BC (IS * IS)               // 4096

typedef unsigned int u32x4 __attribute__((ext_vector_type(4)));
typedef int          i32x8 __attribute__((ext_vector_type(8)));
typedef int          i32x4 __attribute__((ext_vector_type(4)));

__device__ __forceinline__ float tanh_fast(float x) {
#if __has_builtin(__builtin_amdgcn_tanhf)
    return __builtin_amdgcn_tanhf(x);      // v_tanh_f32 (CDNA5 native)
#else
    float e = __expf(-2.0f * fabsf(x));
    float r = (1.0f - e) * __builtin_amdgcn_rcpf(1.0f + e);
    return copysignf(r, x);
#endif
}

__device__ __forceinline__ float sigmoid_fast(float x) {
    return __builtin_amdgcn_rcpf(1.0f + __expf(-x));   // v_exp + v_rcp
}

// t = softplus(10*tanh(v)) / 10
__device__ __forceinline__ float activate(float v) {
    float t = tanh_fast(v);
    return __logf(1.0f + __expf(10.0f * t)) * 0.1f;
}

// ---------------------------------------------------------------------------
// Main kernel (defined first so the disasm snippet shows the TDM op):
// one block per (b,c). TDM DMAs the 16x16 f32 template tile into LDS,
// threads activate it, then each of 256 threads bilinear-samples 16 pixels
// (4x float4 coalesced stores).
// ---------------------------------------------------------------------------
__global__ __launch_bounds__(256) void ACR_render_kernel(const float* __restrict__ pose,
                                                         const float* __restrict__ inten,
                                                         const float* __restrict__ tmpl,
                                                         float* __restrict__ outp) {
    __shared__ float sRaw[TS * TS];
    __shared__ float sAct[TS * TS];

    const int bc  = blockIdx.x;          // 0 .. B*C-1
    const int c   = bc & (NTMPL - 1);
    const int tid = threadIdx.x;

    // ---- TDM: DMA the raw 16x16 f32 tile (1KB) into sRaw -------------------
    if (threadIdx.x < 32) {              // wave 0 only (TDM ignores EXEC)
        uint64_t ga  = (uint64_t)(uintptr_t)(tmpl + (size_t)c * (TS * TS));
        uint32_t lds = (uint32_t)(uintptr_t)(void*)sRaw;   // low 32 bits = LDS offset

        u32x4 g0;
        g0.x = 1u;                                   // count=1, gather off
        g0.y = lds;                                  // lds_addr (bytes)
        g0.z = (uint32_t)ga;                         // global_addr[31:0]
        g0.w = (uint32_t)(ga >> 32) | (2u << 30);    // global_addr[56:32] | type=2

        i32x8 g1;
        g1[0] = (2 << 16);          // workgroup_mask=0, data_size=2 (4 bytes)
        g1[1] = (TS << 16);         // tensor_dim0[15:0]=16 (bits 79:48)
        g1[2] = (TS << 16);         // tensor_dim0 hi=0 | tensor_dim1[15:0]=16
        g1[3] = (TS << 16);         // tensor_dim1 hi=0 | tile_dim0=16
        g1[4] = TS;                 // tile_dim1=16, tile_dim2=0
        g1[5] = TS;                 // tensor_dim0_stride[31:0]=16
        g1[6] = (TS * TS) << 16;    // stride0 hi=0 | tensor_dim1_stride[15:0]=256
        g1[7] = 0;                  // tensor_dim1_stride hi

        i32x4 gz; gz[0] = 0; gz[1] = 0; gz[2] = 0; gz[3] = 0;
#if defined(__clang_major__) && (__clang_major__ >= 23)
        i32x8 gz8;
        gz8[0]=0; gz8[1]=0; gz8[2]=0; gz8[3]=0; gz8[4]=0; gz8[5]=0; gz8[6]=0; gz8[7]=0;
        __builtin_amdgcn_tensor_load_to_lds(g0, g1, gz, gz, gz8, 0);
#else
        __builtin_amdgcn_tensor_load_to_lds(g0, g1, gz, gz, 0);
#endif
        __builtin_amdgcn_s_wait_tensorcnt((short)0);
    }

    // ---- affine params from pose (uniform per block; amortized over 16 px) --
    const float* pp = pose + (size_t)bc * 6;
    float sx = sigmoid_fast(pp[0]) + 1e-2f;
    float sy = sigmoid_fast(pp[1]) + 1e-2f;
    float th = pp[2] * 6.2831853071795864f;
    float sh = tanh_fast(pp[3] * 5.0f);
    float tx = tanh_fast(pp[4] * 5.0f);
    float ty = tanh_fast(pp[5] * 5.0f);
    float cth = __cosf(th), sth = __sinf(th);
    float r00 =  sx * cth + sh * sy * sth;
    float r01 = -sx * sth + sh * sy * cth;
    float r10 =  sy * sth;
    float r11 =  sy * cth;
    // pixel coords: X = 8*(r00*gx + r01*gy + tx) + 7.5  (Win=16, align_corners=False)
    float xa = 8.0f * r00, xb = 8.0f * r01, xc = 8.0f * tx + 7.5f;
    float ya = 8.0f * r10, yb = 8.0f * r11, yc = 8.0f * ty + 7.5f;

    __syncthreads();                     // TDM tile visible to all waves
    sAct[tid] = activate(sRaw[tid]);
    __syncthreads();

    const float I = inten[bc];
    float* ob = outp + T_ELEMS + (size_t)bc * OUT_PER_BC;

    const float GS = 2.0f / (float)IS;             // 0.03125
    const float G0 = 0.5f * GS - 1.0f;             // pixel-center offset

#pragma unroll
    for (int it = 0; it < 4; ++it) {
        int p   = it * 1024 + tid * 4;             // flat pixel index (multiple of 4)
        int py  = p >> 6;
        int px0 = p & 63;
        float gy = (float)py * GS + G0;
        float4 r;
        float* rp = &r.x;
#pragma unroll
        for (int j = 0; j < 4; ++j) {
            float gx = (float)(px0 + j) * GS + G0;
            float X = xa * gx + xb * gy + xc;
            float Y = ya * gx + yb * gy + yc;
            float fx = floorf(X), fy = floorf(Y);
            float wx = X - fx,   wy = Y - fy;
            int ix = (int)fx, iy = (int)fy;
            bool x0v = (unsigned)ix       < (unsigned)TS;
            bool x1v = (unsigned)(ix + 1) < (unsigned)TS;
            bool y0v = (unsigned)iy       < (unsigned)TS;
            bool y1v = (unsigned)(iy + 1) < (unsigned)TS;
            int base = iy * TS + ix;
            float v00 = 0.f, v10 = 0.f, v01 = 0.f, v11 = 0.f;
            if (x0v && y0v) v00 = sAct[base];
            if (x1v && y0v) v10 = sAct[base + 1];
            if (x0v && y1v) v01 = sAct[base + TS];
            if (x1v && y1v) v11 = sAct[base + TS + 1];
            float top = v00 + (v10 - v00) * wx;
            float bot = v01 + (v11 - v01) * wx;
            rp[j] = (top + (bot - top) * wy) * I;
        }
        *(float4*)(ob + p) = r;                    // coalesced global_store_b128
    }
}

// ---------------------------------------------------------------------------
// Kernel 2: activated template output t (16384 elements)
// ---------------------------------------------------------------------------
__global__ __launch_bounds__(256) void ACR_act_kernel(const float* __restrict__ tmpl,
                                                      float* __restrict__ out) {
    int i = blockIdx.x * 256 + threadIdx.x;
    if (i < T_ELEMS) out[i] = activate(tmpl[i]);
}

// ---------------------------------------------------------------------------
extern "C" void kernel_launch(void* const* d_in, const int* in_sizes, int n_in,
                              void* d_out, int out_size, void* d_ws, size_t ws_size,
                              hipStream_t stream) {
    (void)in_sizes; (void)n_in; (void)d_ws; (void)ws_size; (void)out_size;
    const float* pose  = (const float*)d_in[0];   // [B,C,6]
    const float* inten = (const float*)d_in[1];   // [B,C,1]
    const float* tmpl  = (const float*)d_in[2];   // [1,C,16,16]
    float* out = (float*)d_out;                   // [16384 t | 33554432 images]

    ACR_act_kernel<<<(T_ELEMS + 255) / 256, 256, 0, stream>>>(tmpl, out);
    ACR_render_kernel<<<BATCH * NTMPL, 256, 0, stream>>>(pose, inten, tmpl, out);
}